// GTN_17660905521153
// MI455X (gfx1250) — compile-verified
//
#include <hip/hip_runtime.h>
#include <hip/hip_bf16.h>

// ---------------------------------------------------------------------------
// GTN forward for MI455X (gfx1250, wave32).  Dominant work: two batched
// 2048^3 fp32 GEMMs -> V_WMMA_F32_16X16X4_F32 through LDS-staged tiles.
// Workspace use: ~134 MB big buffers + ~15 MB small; everything rewritten
// every call (graph-capture safe, no atomics).
// ---------------------------------------------------------------------------

typedef __attribute__((ext_vector_type(2))) float v2f;
typedef __attribute__((ext_vector_type(4))) float v4f;
typedef __attribute__((ext_vector_type(8))) float v8f;

namespace {
constexpr int  GN   = 2048;                 // graph size N
constexpr long GNN  = (long)GN * GN;        // 2^22
constexpr int  BM   = 128;                  // block tile M
constexpr int  BN   = 64;                   // block tile N
constexpr int  BK   = 16;                   // K step
constexpr int  LSTR = 18;                   // padded LDS row stride (floats)
}

// ---------------------------------------------------------------------------
// Batched fp32 WMMA GEMM:  C[z] = A[z] @ B[z]   (row-major everywhere).
// TRANSA=true means logical A[i][k] = Aphys[k][i] (Aphys row-major, ld=lda).
// Requires M%128==0, N%64==0, K%16==0 (true for every call below).
// ---------------------------------------------------------------------------
template <bool TRANSA>
__global__ __launch_bounds__(256) void gemm_wmma_f32(
    const float* __restrict__ Ag, const float* __restrict__ Bg,
    float* __restrict__ Cg, int M, int Nn, int K, int lda, int ldb, int ldc,
    long sA, long sB, long sC)
{
    __shared__ float As[BM * LSTR];   // [128][16] padded
    __shared__ float Bs[BN * LSTR];   // [64][16]  (N-major, K inner) padded

    const int z = blockIdx.z;
    const float* A = Ag + (long)z * sA;
    const float* B = Bg + (long)z * sB;
    float*       C = Cg + (long)z * sC;

    const int n0   = blockIdx.x * BN;
    const int m0   = blockIdx.y * BM;
    const int tid  = threadIdx.x;
    const int lane = tid & 31;
    const int wave = tid >> 5;        // 0..7
    const int wm   = wave >> 1;       // 0..3  (32-row strip)
    const int wn   = wave & 1;        // 0..1  (32-col strip)
    const int lrow = lane & 15;
    const int lhi  = lane >> 4;

    v8f acc[2][2] = {};

    for (int kb = 0; kb < K; kb += BK) {
        // ---- stage A panel -> As[row][k] ----
        if (!TRANSA) {
            #pragma unroll
            for (int it = 0; it < 2; ++it) {
                int chunk = tid + it * 256;            // 0..511
                int row   = chunk >> 2;                // 0..127
                int c4    = chunk & 3;                 // 0..3
                v4f t = *(const v4f*)(A + (long)(m0 + row) * lda + kb + c4 * 4);
                float* dst = &As[row * LSTR + c4 * 4];
                *(v2f*)(dst)     = (v2f){t.x, t.y};
                *(v2f*)(dst + 2) = (v2f){t.z, t.w};
            }
        } else {
            #pragma unroll
            for (int it = 0; it < 2; ++it) {
                int chunk = tid + it * 256;            // 0..511
                int pr    = chunk >> 5;                // 0..15 (k offset)
                int seg   = chunk & 31;                // 0..31 (4 logical rows)
                v4f t = *(const v4f*)(A + (long)(kb + pr) * lda + m0 + seg * 4);
                As[(seg * 4 + 0) * LSTR + pr] = t.x;
                As[(seg * 4 + 1) * LSTR + pr] = t.y;
                As[(seg * 4 + 2) * LSTR + pr] = t.z;
                As[(seg * 4 + 3) * LSTR + pr] = t.w;
            }
        }
        // ---- stage B panel -> Bs[n][k] (transpose 16x64 block) ----
        {
            int r  = tid >> 4;                          // 0..15 (k)
            int c4 = tid & 15;                          // 0..15 (4 cols)
            v4f t = *(const v4f*)(B + (long)(kb + r) * ldb + n0 + c4 * 4);
            Bs[(c4 * 4 + 0) * LSTR + r] = t.x;
            Bs[(c4 * 4 + 1) * LSTR + r] = t.y;
            Bs[(c4 * 4 + 2) * LSTR + r] = t.z;
            Bs[(c4 * 4 + 3) * LSTR + r] = t.w;
        }
        __syncthreads();

        // ---- 4 wmma K-substeps of 4 ----
        #pragma unroll
        for (int kk = 0; kk < BK; kk += 4) {
            v2f af[2], bf[2];
            #pragma unroll
            for (int ti = 0; ti < 2; ++ti) {
                int mo = wm * 32 + ti * 16;
                // A frag (16x4 f32): lane M=lrow, VGPR0:K=kk+lhi*2, VGPR1:K+1
                af[ti] = *(const v2f*)&As[(mo + lrow) * LSTR + kk + lhi * 2];
            }
            #pragma unroll
            for (int tj = 0; tj < 2; ++tj) {
                int no = wn * 32 + tj * 16;
                // B frag (4x16 f32): lane N=lrow, VGPR0:K=kk+lhi*2, VGPR1:K+1
                bf[tj] = *(const v2f*)&Bs[(no + lrow) * LSTR + kk + lhi * 2];
            }
            #pragma unroll
            for (int ti = 0; ti < 2; ++ti)
                #pragma unroll
                for (int tj = 0; tj < 2; ++tj)
                    acc[ti][tj] = __builtin_amdgcn_wmma_f32_16x16x4_f32(
                        false, af[ti], false, bf[tj], (short)0, acc[ti][tj],
                        false, false);
        }
        __syncthreads();
    }

    // ---- store: VGPR r -> rows (r, r+8); col = lane%16 ----
    #pragma unroll
    for (int ti = 0; ti < 2; ++ti) {
        #pragma unroll
        for (int tj = 0; tj < 2; ++tj) {
            int mo = m0 + wm * 32 + ti * 16;
            int no = n0 + wn * 32 + tj * 16;
            #pragma unroll
            for (int r = 0; r < 8; ++r)
                C[(long)(mo + r + lhi * 8) * ldc + no + lrow] = acc[ti][tj][r];
        }
    }
}

// ---------------------------------------------------------------------------
// softmax over E=5 for w0a/w0b/w1a ([2][5] each) -> ws (sm) and d_out Ws
// ---------------------------------------------------------------------------
__global__ void softmax_kernel(const float* __restrict__ w0a,
                               const float* __restrict__ w0b,
                               const float* __restrict__ w1a,
                               float* __restrict__ sm, float* __restrict__ outWs)
{
    int t = threadIdx.x;
    if (t >= 6) return;
    const float* src = (t < 2) ? w0a : (t < 4) ? w0b : w1a;
    int c = t & 1;
    const float* row = src + c * 5;
    float m = row[0];
    for (int e = 1; e < 5; ++e) m = fmaxf(m, row[e]);
    float ex[5], s = 0.f;
    for (int e = 0; e < 5; ++e) { ex[e] = __expf(row[e] - m); s += ex[e]; }
    int base = (t >> 1) * 10 + c * 5;
    for (int e = 0; e < 5; ++e) {
        float v = ex[e] / s;
        sm[base + e]    = v;
        outWs[base + e] = v;
    }
}

// ---------------------------------------------------------------------------
// gtconv: a/b/a1[c][n][m] = sum_e sm[..][c][e] * A[n][m][e]   (E=5 inner)
// ---------------------------------------------------------------------------
__global__ void gtconv_kernel(const float* __restrict__ A,
                              const float* __restrict__ sm,
                              float* __restrict__ a, float* __restrict__ b,
                              float* __restrict__ a1)
{
    int idx = blockIdx.x * blockDim.x + threadIdx.x;   // < GN*GN
    float v[5];
    #pragma unroll
    for (int e = 0; e < 5; ++e) v[e] = A[(long)idx * 5 + e];
    #pragma unroll
    for (int c = 0; c < 2; ++c) {
        float sa = 0.f, sb = 0.f, s1 = 0.f;
        #pragma unroll
        for (int e = 0; e < 5; ++e) {
            sa += sm[c * 5 + e]      * v[e];
            sb += sm[10 + c * 5 + e] * v[e];
            s1 += sm[20 + c * 5 + e] * v[e];
        }
        a [(long)c * GNN + idx] = sa;
        b [(long)c * GNN + idx] = sb;
        a1[(long)c * GNN + idx] = s1;
    }
}

// ---------------------------------------------------------------------------
// Column sums of zero-diag(H) (+1 if addOne); inv = (s==0)?0:1/s; opt diag out
// ---------------------------------------------------------------------------
__global__ void colsum_kernel(const float* __restrict__ Hm,
                              float* __restrict__ inv, float* __restrict__ diag,
                              int addOne, long strideM, int strideV)
{
    int c = blockIdx.y;
    int j = blockIdx.x * blockDim.x + threadIdx.x;     // column
    const float* Hc = Hm + (long)c * strideM;
    float s = 0.f;
    for (int i = 0; i < GN; ++i) s += Hc[(long)i * GN + j];
    float d = Hc[(long)j * GN + j];
    s -= d;
    if (addOne) s += 1.f;
    float iv = (s == 0.f) ? 0.f : 1.f / s;
    inv[c * strideV + j] = iv;
    if (diag) diag[c * strideV + j] = d;
}

// In-place Hn = norm(H1, add=false): H1[c][i][j] = (i==j?0:H1)*inv1[c][j]
__global__ void hn_scale_kernel(float* __restrict__ H, const float* __restrict__ inv)
{
    int idx = blockIdx.x * blockDim.x + threadIdx.x;   // < 2*GNN (fits int)
    int c = idx >> 22;
    int i = (idx >> 11) & (GN - 1);
    int j = idx & (GN - 1);
    float v = H[idx];
    H[idx] = (i == j) ? 0.f : v * inv[(c << 11) + j];
}

// Hsum = H2[0] + H2[1]
__global__ void hsum_kernel(const float* __restrict__ H2, float* __restrict__ Hs)
{
    int p = blockIdx.x * blockDim.x + threadIdx.x;     // < GNN
    Hs[p] = H2[p] + H2[GNN + p];
}

// X_cat[i][c*128+d] = relu((G[c][i][d] + (1-diag2)*XW[i][d]) * inv2[c][i])
__global__ void xcat_kernel(const float* __restrict__ G, const float* __restrict__ XW,
                            const float* __restrict__ inv2, const float* __restrict__ diag2,
                            float* __restrict__ Xcat)
{
    int idx = blockIdx.x * blockDim.x + threadIdx.x;   // < 2*2048*128
    int c = idx >> 18;
    int r = idx & ((1 << 18) - 1);
    int i = r >> 7;
    int d = r & 127;
    float v = (G[idx] + (1.f - diag2[(c << 11) + i]) * XW[(i << 7) + d])
              * inv2[(c << 11) + i];
    Xcat[(i << 8) + (c << 7) + d] = fmaxf(v, 0.f);
}

// Xout[i][d] = (G[i][d] + (1-diagS[i])*Y[i][d]) * invS[i]
__global__ void head_norm_kernel(const float* __restrict__ G, const float* __restrict__ Y,
                                 const float* __restrict__ invS, const float* __restrict__ diagS,
                                 float* __restrict__ Xout, int width)
{
    int idx = blockIdx.x * blockDim.x + threadIdx.x;   // < 2048*width
    int i = idx / width;
    Xout[idx] = (G[idx] + (1.f - diagS[i]) * Y[idx]) * invS[i];
}

// y[t][k] = X_o[target_x[t]] . lin_w[k] + lin_b[k]
__global__ void classifier_kernel(const float* __restrict__ Xo, const int* __restrict__ tx,
                                  const float* __restrict__ lw, const float* __restrict__ lb,
                                  float* __restrict__ y)
{
    int idx = blockIdx.x * blockDim.x + threadIdx.x;   // < 1024*8
    int t = idx >> 3;
    int k = idx & 7;
    const float* xr = Xo + tx[t] * 128;
    const float* wr = lw + k * 128;
    float s = 0.f;
    #pragma unroll 8
    for (int d = 0; d < 128; ++d) s += xr[d] * wr[d];
    y[idx] = s + lb[k];
}

// BCE partial sums over 2048*256 elems; 256 blocks x 2048 elems, LDS reduce
__global__ void lossR_kernel(const float* __restrict__ Xr, const float* __restrict__ X,
                             float* __restrict__ partial)
{
    __shared__ float red[256];
    int b = blockIdx.x, t = threadIdx.x;
    float s = 0.f;
    int base = b * 2048;
    for (int k = t; k < 2048; k += 256) {
        int idx = base + k;
        float v = Xr[idx];
        if (v > 1.f || v < 0.f) v = 1.f / (1.f + __expf(-v));
        float p = fminf(fmaxf(v, 1e-7f), 1.f - 1e-7f);
        float x = X[idx];
        s += x * __logf(p) + (1.f - x) * log1pf(-p);
    }
    red[t] = s;
    __syncthreads();
    for (int off = 128; off > 0; off >>= 1) {
        if (t < off) red[t] += red[t + off];
        __syncthreads();
    }
    if (t == 0) partial[b] = red[0];
}

// cross-entropy partial sums over 1024 rows of 8 logits; 4 blocks
__global__ void lossC_kernel(const float* __restrict__ y, const int* __restrict__ tgt,
                             float* __restrict__ partial)
{
    __shared__ float red[256];
    int b = blockIdx.x, t = threadIdx.x;
    int row = b * 256 + t;                 // < 1024
    const float* yr = y + row * 8;
    float m = yr[0];
    #pragma unroll
    for (int k = 1; k < 8; ++k) m = fmaxf(m, yr[k]);
    float se = 0.f;
    #pragma unroll
    for (int k = 0; k < 8; ++k) se += __expf(yr[k] - m);
    float lse = m + __logf(se);
    red[t] = lse - yr[tgt[row]];           // = -log_softmax at target
    __syncthreads();
    for (int off = 128; off > 0; off >>= 1) {
        if (t < off) red[t] += red[t + off];
        __syncthreads();
    }
    if (t == 0) partial[b] = red[0];
}

__global__ void finalize_kernel(const float* __restrict__ pR, const float* __restrict__ pC,
                                float* __restrict__ out)
{
    if (threadIdx.x == 0) {
        float sR = 0.f;
        for (int i = 0; i < 256; ++i) sR += pR[i];
        float sC = 0.f;
        for (int i = 0; i < 4; ++i) sC += pC[i];
        float lossR = -sR / 524288.f;      // mean over 2048*256
        float lossC = sC / 1024.f;
        out[0]    = lossC + lossR;
        out[8223] = lossR;
    }
}

// ---------------------------------------------------------------------------
extern "C" void kernel_launch(void* const* d_in, const int* in_sizes, int n_in,
                              void* d_out, int out_size, void* d_ws, size_t ws_size,
                              hipStream_t stream)
{
    (void)in_sizes; (void)n_in; (void)out_size; (void)ws_size;
    const float* A    = (const float*)d_in[0];   // [N][N][E]
    const float* X    = (const float*)d_in[1];   // [N][256]
    const float* W    = (const float*)d_in[2];   // [256][128]
    const float* W1   = (const float*)d_in[3];   // [256][128]
    const float* W2   = (const float*)d_in[4];   // [256][256]
    const float* linw = (const float*)d_in[5];   // [8][128]
    const float* linb = (const float*)d_in[6];   // [8]
    const float* w0a  = (const float*)d_in[7];
    const float* w0b  = (const float*)d_in[8];
    const float* w1a  = (const float*)d_in[9];
    const int*   tx   = (const int*)d_in[10];    // [1024]
    const int*   tgt  = (const int*)d_in[11];    // [1024]
    float* out = (float*)d_out;                  // 1 + 8192 + 30 + 1
    float* ws  = (float*)d_ws;                   // needs ~134.4 MB

    // ---- workspace layout (floats) ----
    float* bufA = ws;                 // [2][N][N]: a       -> H2
    float* bufB = ws + 2 * GNN;       // [2][N][N]: b       -> Hsum + small mats
    float* bufC = ws + 4 * GNN;       // [2][N][N]: a1
    float* bufH = ws + 6 * GNN;       // [2][N][N]: H1 -> Hn (in place)
    float* smallb = ws + 8 * GNN;
    float* sm    = smallb;            // 30 (+pad)
    float* inv1  = smallb + 32;       // [2][2048]
    float* inv2  = inv1 + 2 * GN;     // [2][2048]
    float* diag2 = inv2 + 2 * GN;     // [2][2048]
    float* invS  = diag2 + 2 * GN;    // [2048]
    float* diagS = invS + GN;         // [2048]
    float* pR    = diagS + GN;        // [256]
    float* pC    = pR + 256;          // [4]

    float* Hsum = bufB;                                   // [N][N]
    float* o    = bufB + GNN;                             // fits: 3.7M < GNN
    float* XW   = o;  o += 2048 * 128;                    // X @ W
    float* Xcat = o;  o += 2048 * 256;
    float* G    = o;  o += 2L * 2048 * 128;               // H2^T @ XW
    float* Y2   = o;  o += 2048 * 256;                    // Xcat @ W2
    float* Y1   = o;  o += 2048 * 128;                    // Xcat @ W1
    float* GR   = o;  o += 2048 * 256;                    // Hsum^T @ Y2
    float* GO   = o;  o += 2048 * 128;                    // Hsum^T @ Y1
    float* Xr   = o;  o += 2048 * 256;
    float* Xo   = o;  o += 2048 * 128;

    const int NT = (int)(GNN / 256);                      // elemwise grid helper

    // 1) edge-type softmaxes (also written straight into out[8193..8222])
    softmax_kernel<<<1, 32, 0, stream>>>(w0a, w0b, w1a, sm, out + 1 + 8192);

    // 2) a, b, a1 from A in one pass
    gtconv_kernel<<<NT, 256, 0, stream>>>(A, sm, bufA, bufB, bufC);

    // 3) H1[c] = a[c] @ b[c]
    gemm_wmma_f32<false><<<dim3(GN / BN, GN / BM, 2), 256, 0, stream>>>(
        bufA, bufB, bufH, GN, GN, GN, GN, GN, GN, GNN, GNN, GNN);

    // 4) layer-1 norm (add=false), in place: H1 -> Hn
    colsum_kernel<<<dim3(GN / 256, 2), 256, 0, stream>>>(bufH, inv1, nullptr, 0, GNN, GN);
    hn_scale_kernel<<<2 * NT, 256, 0, stream>>>(bufH, inv1);

    // 5) H2[c] = Hn[c] @ a1[c]
    gemm_wmma_f32<false><<<dim3(GN / BN, GN / BM, 2), 256, 0, stream>>>(
        bufH, bufC, bufA, GN, GN, GN, GN, GN, GN, GNN, GNN, GNN);

    // 6) XW = X @ W   [2048 x 128]
    gemm_wmma_f32<false><<<dim3(128 / BN, GN / BM, 1), 256, 0, stream>>>(
        X, W, XW, GN, 128, 256, 256, 128, 128, 0, 0, 0);

    // 7) per-channel GCN: colsums(+1)/diag of H2, G = H2^T @ XW, fuse epilogue
    colsum_kernel<<<dim3(GN / 256, 2), 256, 0, stream>>>(bufA, inv2, diag2, 1, GNN, GN);
    gemm_wmma_f32<true><<<dim3(128 / BN, GN / BM, 2), 256, 0, stream>>>(
        bufA, XW, G, GN, 128, GN, GN, 128, 128, GNN, 0, 2048L * 128);
    xcat_kernel<<<2 * 2048 * 128 / 256, 256, 0, stream>>>(G, XW, inv2, diag2, Xcat);

    // 8) Hsum = H2[0]+H2[1]; its norm stats
    hsum_kernel<<<NT, 256, 0, stream>>>(bufA, Hsum);
    colsum_kernel<<<dim3(GN / 256, 1), 256, 0, stream>>>(Hsum, invS, diagS, 1, 0, GN);

    // 9) Y2 = Xcat@W2, Y1 = Xcat@W1
    gemm_wmma_f32<false><<<dim3(256 / BN, GN / BM, 1), 256, 0, stream>>>(
        Xcat, W2, Y2, GN, 256, 256, 256, 256, 256, 0, 0, 0);
    gemm_wmma_f32<false><<<dim3(128 / BN, GN / BM, 1), 256, 0, stream>>>(
        Xcat, W1, Y1, GN, 128, 256, 256, 128, 128, 0, 0, 0);

    // 10) GR = Hsum^T @ Y2, GO = Hsum^T @ Y1, then normalized heads
    gemm_wmma_f32<true><<<dim3(256 / BN, GN / BM, 1), 256, 0, stream>>>(
        Hsum, Y2, GR, GN, 256, GN, GN, 256, 256, 0, 0, 0);
    gemm_wmma_f32<true><<<dim3(128 / BN, GN / BM, 1), 256, 0, stream>>>(
        Hsum, Y1, GO, GN, 128, GN, GN, 128, 128, 0, 0, 0);
    head_norm_kernel<<<2048 * 256 / 256, 256, 0, stream>>>(GR, Y2, invS, diagS, Xr, 256);
    head_norm_kernel<<<2048 * 128 / 256, 256, 0, stream>>>(GO, Y1, invS, diagS, Xo, 128);

    // 11) classifier logits -> out[1..8192]
    classifier_kernel<<<1024 * 8 / 256, 256, 0, stream>>>(Xo, tx, linw, linb, out + 1);

    // 12) losses (deterministic two-stage reductions)
    lossR_kernel<<<256, 256, 0, stream>>>(Xr, X, pR);
    lossC_kernel<<<4, 256, 0, stream>>>(out + 1, tgt, pC);
    finalize_kernel<<<1, 32, 0, stream>>>(pR, pC, out);
}